// MovingPoints_16716012716047
// MI455X (gfx1250) — compile-verified
//
#include <hip/hip_runtime.h>

typedef __attribute__((ext_vector_type(2))) float v2f;
typedef __attribute__((ext_vector_type(8))) float v8f;

#define TILE 16

// One wave32 owns one 16-row tile (ti) and a contiguous chunk of 16-col tiles.
// WMMA computes the full 16x16 squared-distance tile via the K=4 encoding:
//   A[m] = (x_i, y_i, n_i, 1),  B[:,n] = (-2 x_j, -2 y_j, 1, n_j)
//   => D[m][n] = n_i + n_j - 2(x_i x_j + y_i y_j) = |p_i - p_j|^2
__global__ __launch_bounds__(256) void mp_pair_kernel(
    const float* __restrict__ x, const float* __restrict__ y,
    float* __restrict__ partial, int row_tiles, int chunks)
{
  const int lane   = threadIdx.x & 31;
  const int waveIb = threadIdx.x >> 5;
  const int w      = blockIdx.x * (blockDim.x >> 5) + waveIb;
  const int nwaves = row_tiles * chunks;
  if (w >= nwaves) return;                 // whole-wave uniform: EXEC stays all-1s

  const int ti       = w / chunks;
  const int chunk    = w % chunks;
  const int cols_per = row_tiles / chunks;
  const int tj0      = chunk * cols_per;

  const int  lane16 = lane & 15;
  const bool hi     = lane >= 16;
  const int  rbase  = ti * TILE;

  // ---- A fragment (loop-invariant). 32-bit A 16x4 layout:
  // lanes 0-15: VGPR0=K0, VGPR1=K1 ; lanes 16-31: VGPR0=K2, VGPR1=K3 ; M = lane&15
  const float xi  = x[rbase + lane16];
  const float yi  = y[rbase + lane16];
  const float niv = __builtin_fmaf(xi, xi, yi * yi);
  v2f a;
  a.x = hi ? niv  : xi;    // K2 = n_i   | K0 = x_i
  a.y = hi ? 1.0f : yi;    // K3 = 1     | K1 = y_i

  // Row coordinates needed by the epilogue: D VGPR r holds row m = r + 8*hi.
  const int ro = rbase + (hi ? 8 : 0);
  float xi8[8], yi8[8];
#pragma unroll
  for (int r = 0; r < 8; ++r) { xi8[r] = x[ro + r]; yi8[r] = y[ro + r]; }

  float sum = 0.0f;
  for (int tj = tj0; tj < tj0 + cols_per; ++tj) {
    const int   cbase = tj * TILE;
    const float xj    = x[cbase + lane16];
    const float yj    = y[cbase + lane16];
    const float njv   = __builtin_fmaf(xj, xj, yj * yj);

    // ---- B fragment. B 4x16: col N = lane&15;
    // lanes 0-15: VGPR0=K0, VGPR1=K1 ; lanes 16-31: VGPR0=K2, VGPR1=K3
    v2f b;
    b.x = hi ? 1.0f : -2.0f * xj;   // K2 = 1    | K0 = -2 x_j
    b.y = hi ? njv  : -2.0f * yj;   // K3 = n_j  | K1 = -2 y_j

    v8f c = {0.f, 0.f, 0.f, 0.f, 0.f, 0.f, 0.f, 0.f};
    v8f d = __builtin_amdgcn_wmma_f32_16x16x4_f32(
        /*neg_a=*/false, a, /*neg_b=*/false, b,
        /*c_mod=*/(short)0, c, /*reuse_a=*/false, /*reuse_b=*/false);

    const int gj = cbase + lane16;
#pragma unroll
    for (int r = 0; r < 8; ++r) {
      float sq = d[r];
      // Cancellation guard: for close pairs recompute sq exactly (reference path).
      const float dx  = xi8[r] - xj;
      const float dy  = yi8[r] - yj;
      const float sqe = __builtin_fmaf(dx, dx, dy * dy);
      sq = (sq < 1e-2f) ? sqe : sq;
      const float inv = __builtin_amdgcn_rsqf(sq);   // v_rsq_f32
      // Diagonal: sq==0 -> inv=+inf, masked to 0 (matches 1/inf = 0 in reference).
      sum += ((ro + r) == gj) ? 0.0f : inv;
    }
  }

  // wave32 butterfly reduce
#pragma unroll
  for (int off = 16; off > 0; off >>= 1)
    sum += __shfl_xor(sum, off, 32);
  if (lane == 0) partial[w] = sum;
}

// Deterministic final reduction: fixed-order sums + LDS tree; also adds KE.
__global__ __launch_bounds__(256) void mp_reduce_kernel(
    const float* __restrict__ partial, int nparts,
    const float* __restrict__ px, const float* __restrict__ py, int n,
    float* __restrict__ out)
{
  __shared__ float red[256];
  const int t = threadIdx.x;
  float s = 0.0f;
  for (int i = t; i < nparts; i += 256) s += partial[i];
  for (int i = t; i < n; i += 256) {
    const float a = px[i], b = py[i];
    s += 0.5f * __builtin_fmaf(a, a, b * b);   // 1/(2*MASS) = 0.5
  }
  red[t] = s;
  __syncthreads();
  for (int stride = 128; stride > 0; stride >>= 1) {
    if (t < stride) red[t] += red[t + stride];
    __syncthreads();
  }
  if (t == 0) out[0] = red[0];
}

extern "C" void kernel_launch(void* const* d_in, const int* in_sizes, int n_in,
                              void* d_out, int out_size, void* d_ws, size_t ws_size,
                              hipStream_t stream) {
  const float* x  = (const float*)d_in[0];
  const float* y  = (const float*)d_in[1];
  const float* px = (const float*)d_in[2];
  const float* py = (const float*)d_in[3];
  float* out      = (float*)d_out;
  float* partial  = (float*)d_ws;

  const int n         = in_sizes[0];          // 16384
  const int row_tiles = n / TILE;             // 1024
  const int chunks    = (row_tiles % 8 == 0) ? 8 : 1;
  const int nwaves    = row_tiles * chunks;   // 8192 -> 32 KB of d_ws
  const int blocks    = (nwaves + 7) / 8;     // 8 waves per 256-thread block

  mp_pair_kernel<<<blocks, 256, 0, stream>>>(x, y, partial, row_tiles, chunks);
  mp_reduce_kernel<<<1, 256, 0, stream>>>(partial, nwaves, px, py, n, out);
}